// Ternarize_66743791780424
// MI455X (gfx1250) — compile-verified
//
#include <hip/hip_runtime.h>
#include <stdint.h>

// Problem geometry (fixed by reference: 8192 x 8192 fp32).
#define ROW_N   8192
#define BLOCK   256
#define NWAVES  (BLOCK / 32)            // 8 waves (wave32)
#define PER_T   (ROW_N / BLOCK)         // 32 floats per thread
#define CHUNKS  (PER_T / 4)             // 8 x float4 per thread

typedef float v4f __attribute__((ext_vector_type(4)));
typedef int   v4i __attribute__((ext_vector_type(4)));
typedef __attribute__((address_space(1))) v4i* as1_v4i;
typedef __attribute__((address_space(3))) void* as3_void;
typedef __attribute__((address_space(3))) v4i* as3_v4i;

// CDNA5 async global->LDS copy (ASYNCcnt-tracked, bypasses VGPRs).
__device__ __forceinline__ void async_copy_b128(const float* gsrc, float* ldst) {
#if __has_builtin(__builtin_amdgcn_global_load_async_to_lds_b128)
    __builtin_amdgcn_global_load_async_to_lds_b128(
        (as1_v4i)(uintptr_t)gsrc,        // global src: int4 addrspace(1)*
        (as3_v4i)(as3_void)ldst,         // LDS dst:    int4 addrspace(3)*
        /*offset=*/0, /*cpol=*/0);
#else
    uint32_t loff = (uint32_t)(uintptr_t)(as3_void)ldst;  // LDS byte offset
    asm volatile("global_load_async_to_lds_b128 %0, %1, off"
                 :: "v"(loff), "v"(gsrc) : "memory");
#endif
}

__device__ __forceinline__ void wait_async0() {
#if __has_builtin(__builtin_amdgcn_s_wait_asynccnt)
    __builtin_amdgcn_s_wait_asynccnt(0);
#else
    asm volatile("s_wait_asynccnt 0x0" ::: "memory");
#endif
}

__device__ __forceinline__ float wave_sum(float v) {
    #pragma unroll
    for (int m = 16; m >= 1; m >>= 1) v += __shfl_xor(v, m, 32);
    return v;
}

__global__ __launch_bounds__(BLOCK)
void Ternarize_66743791780424_kernel(const float* __restrict__ x,
                                     float* __restrict__ out) {
    __shared__ __align__(16) float row[ROW_N];   // 32 KB staged row
    __shared__ float redA[NWAVES];
    __shared__ float redB[NWAVES];

    const int tid = threadIdx.x;
    const int wid = tid >> 5;
    const size_t rbase = (size_t)blockIdx.x * ROW_N;
    const float* __restrict__ xrow = x + rbase;
    float* __restrict__ orow = out + rbase;

    // ---- Stage: async global -> LDS, one HBM read of the row ----
    #pragma unroll
    for (int c = 0; c < CHUNKS; ++c) {
        const int idx = c * (BLOCK * 4) + tid * 4;
        async_copy_b128(xrow + idx, &row[idx]);
    }
    wait_async0();          // this wave's async copies landed in LDS
    __syncthreads();        // all waves' copies visible workgroup-wide

    // ---- Pass 1 (LDS): sum |x| -> delta ----
    float s0 = 0.0f;
    #pragma unroll
    for (int c = 0; c < CHUNKS; ++c) {
        const int idx = c * (BLOCK * 4) + tid * 4;
        v4f v = *reinterpret_cast<const v4f*>(&row[idx]);
        s0 += __builtin_fabsf(v.x) + __builtin_fabsf(v.y)
            + __builtin_fabsf(v.z) + __builtin_fabsf(v.w);
    }
    s0 = wave_sum(s0);
    if ((tid & 31) == 0) redA[wid] = s0;
    __syncthreads();
    float total = 0.0f;
    #pragma unroll
    for (int w = 0; w < NWAVES; ++w) total += redA[w];
    const float delta = 0.75f * total * (1.0f / (float)ROW_N);
    __syncthreads();   // protect redA/redB before reuse

    // ---- Pass 2 (LDS): sum |x| over |x|>delta, and count -> alpha ----
    float s1 = 0.0f, cnt = 0.0f;
    #pragma unroll
    for (int c = 0; c < CHUNKS; ++c) {
        const int idx = c * (BLOCK * 4) + tid * 4;
        v4f v = *reinterpret_cast<const v4f*>(&row[idx]);
        float a;
        a = __builtin_fabsf(v.x); { float m = (a > delta) ? 1.0f : 0.0f; s1 += a * m; cnt += m; }
        a = __builtin_fabsf(v.y); { float m = (a > delta) ? 1.0f : 0.0f; s1 += a * m; cnt += m; }
        a = __builtin_fabsf(v.z); { float m = (a > delta) ? 1.0f : 0.0f; s1 += a * m; cnt += m; }
        a = __builtin_fabsf(v.w); { float m = (a > delta) ? 1.0f : 0.0f; s1 += a * m; cnt += m; }
    }
    s1  = wave_sum(s1);
    cnt = wave_sum(cnt);
    if ((tid & 31) == 0) { redA[wid] = s1; redB[wid] = cnt; }
    __syncthreads();
    float abssum = 0.0f, count = 0.0f;
    #pragma unroll
    for (int w = 0; w < NWAVES; ++w) { abssum += redA[w]; count += redB[w]; }
    const float alpha = abssum / count;

    // ---- Pass 3 (LDS): ternarize + streaming (non-temporal) store ----
    #pragma unroll
    for (int c = 0; c < CHUNKS; ++c) {
        const int idx = c * (BLOCK * 4) + tid * 4;
        v4f v = *reinterpret_cast<const v4f*>(&row[idx]);
        v4f o;
        o.x = (v.x > delta) ? alpha : ((v.x < -delta) ? -alpha : 0.0f);
        o.y = (v.y > delta) ? alpha : ((v.y < -delta) ? -alpha : 0.0f);
        o.z = (v.z > delta) ? alpha : ((v.z < -delta) ? -alpha : 0.0f);
        o.w = (v.w > delta) ? alpha : ((v.w < -delta) ? -alpha : 0.0f);
        __builtin_nontemporal_store(o, reinterpret_cast<v4f*>(&orow[idx]));
    }
}

extern "C" void kernel_launch(void* const* d_in, const int* in_sizes, int n_in,
                              void* d_out, int out_size, void* d_ws, size_t ws_size,
                              hipStream_t stream) {
    const float* x = (const float*)d_in[0];
    float* out = (float*)d_out;
    const int rows = in_sizes[0] / ROW_N;   // 8192 rows of 8192 fp32
    Ternarize_66743791780424_kernel<<<dim3(rows), dim3(BLOCK), 0, stream>>>(x, out);
}